// FeatureAttention_39771397161379
// MI455X (gfx1250) — compile-verified
//
#include <hip/hip_runtime.h>
#include <math.h>

#define BB 64
#define FF 2048

typedef float v2f __attribute__((ext_vector_type(2)));
typedef float v8f __attribute__((ext_vector_type(8)));

#if defined(__has_builtin)
#if __has_builtin(__builtin_amdgcn_exp2f)
#define EXP2F(x) __builtin_amdgcn_exp2f(x)
#else
#define EXP2F(x) exp2f(x)
#endif
#else
#define EXP2F(x) exp2f(x)
#endif

// ---------------------------------------------------------------------------
// Kernel 1: Q/K/V = x @ W^T via V_WMMA_F32_16X16X4_F32.
// One wave (32 threads) owns one 16x16 output tile; blockIdx.z picks Wq/Wk/Wv.
// A fragment (16x4 f32, 2 VGPR): lane&15 = M row, lanes 16-31 hold K+2 half.
// B fragment (4x16 f32, 2 VGPR): lane&15 = N col (from W^T, i.e. W row n0+N).
// C/D (16x16 f32, 8 VGPR): lane&15 = N, VGPR r = M (lanes 16-31: M = 8+r).
// ---------------------------------------------------------------------------
__global__ __launch_bounds__(32) void qkv_wmma_gemm(
    const float* __restrict__ x,
    const float* __restrict__ Wq,
    const float* __restrict__ Wk,
    const float* __restrict__ Wv,
    float* __restrict__ Q,
    float* __restrict__ K,
    float* __restrict__ V)
{
    const int lane = threadIdx.x;          // 0..31
    const int n0   = blockIdx.x * 16;      // feature tile (output col)
    const int m0   = blockIdx.y * 16;      // batch tile  (output row)

    const float* W;
    float*       O;
    if (blockIdx.z == 0)      { W = Wq; O = Q; }
    else if (blockIdx.z == 1) { W = Wk; O = K; }
    else                      { W = Wv; O = V; }

    const int r15   = lane & 15;
    const int khalf = (lane >> 4) << 1;    // lanes 0-15: K=0,1 ; lanes 16-31: K=2,3
    const float* arow = x + (size_t)(m0 + r15) * FF + khalf;   // A: x rows
    const float* brow = W + (size_t)(n0 + r15) * FF + khalf;   // B: W^T cols = W rows

    v8f acc = {};
    for (int kk = 0; kk < FF; kk += 32) {
        __builtin_prefetch(brow + kk + 512, 0, 1);   // global_prefetch_b8, ~2KB ahead
#pragma unroll
        for (int u = 0; u < 8; ++u) {
            const int kc = kk + u * 4;
            v2f a = *(const v2f*)(arow + kc);
            v2f b = *(const v2f*)(brow + kc);
            acc = __builtin_amdgcn_wmma_f32_16x16x4_f32(
                false, a, false, b, (short)0, acc, false, false);
        }
    }

    const int mbase = m0 + ((lane >> 4) << 3);   // lanes 0-15 -> M=r, 16-31 -> M=8+r
#pragma unroll
    for (int r = 0; r < 8; ++r)
        O[(size_t)(mbase + r) * FF + n0 + r15] = acc[r];
}

// ---------------------------------------------------------------------------
// Kernel 2: fused rank-1 softmax + weighted sum.
// out[b,f] = sum_g softmax_g( mask? q[b,f]*k[b,g] : -inf ) * v[b,g]
// grid = (F/512, B); block = 256 threads, 2 f-values per thread.
// k,v staged as interleaved float2 in LDS (one ds_load_b64 broadcast per g);
// mask packed to bits via LDS atomicOr; kmin/kmax tree-reduced in LDS.
// ---------------------------------------------------------------------------
__global__ __launch_bounds__(256) void feature_attn(
    const float* __restrict__ Q,
    const float* __restrict__ K,
    const float* __restrict__ V,
    const int*   __restrict__ mask,
    float*       __restrict__ out)
{
    __shared__ float2   kv[FF];          // 16 KB: {k, v} interleaved
    __shared__ unsigned mbits[FF / 32];  // 256 B: packed mask bits
    __shared__ float    rmax[256];
    __shared__ float    rmin[256];

    const int b   = blockIdx.y;
    const int tid = threadIdx.x;

    if (tid < FF / 32) mbits[tid] = 0u;
    __syncthreads();

    // Stage k/v into LDS, pack mask bits, fold per-thread kmin/kmax (unmasked only).
    float lmax = -3.402823466e38f, lmin = 3.402823466e38f;
    for (int g = tid; g < FF; g += 256) {
        const float kg = K[(size_t)b * FF + g];
        const float vg = V[(size_t)b * FF + g];
        const int   mg = mask[(size_t)b * FF + g];
        kv[g] = make_float2(kg, vg);
        if (mg != 0) {
            atomicOr(&mbits[g >> 5], 1u << (g & 31));   // ds_or_b32
            lmax = fmaxf(lmax, kg);
            lmin = fminf(lmin, kg);
        }
    }
    rmax[tid] = lmax;
    rmin[tid] = lmin;
    __syncthreads();

#pragma unroll
    for (int s = 128; s > 0; s >>= 1) {
        if (tid < s) {
            rmax[tid] = fmaxf(rmax[tid], rmax[tid + s]);
            rmin[tid] = fminf(rmin[tid], rmin[tid + s]);
        }
        __syncthreads();
    }
    const float kmax = rmax[0];
    const float kmin = rmin[0];

    // Two f-values per thread, all math in the exp2 domain.
    const int   f0 = blockIdx.x * 512;
    const int   fa = f0 + tid;
    const int   fb = fa + 256;
    const float L2E = 1.44269504088896340736f;
    const float t0 = Q[(size_t)b * FF + fa] * L2E;
    const float t1 = Q[(size_t)b * FF + fb] * L2E;
    const float m0 = (t0 >= 0.f) ? t0 * kmax : t0 * kmin;  // row max (stability)
    const float m1 = (t1 >= 0.f) ? t1 * kmax : t1 * kmin;

    float den0 = 0.f, num0 = 0.f, den1 = 0.f, num1 = 0.f;
    for (int gw = 0; gw < FF / 32; ++gw) {
        const unsigned mb = mbits[gw];
#pragma unroll
        for (int gg = 0; gg < 32; ++gg) {
            const float2 kvv = kv[gw * 32 + gg];
            const bool   on  = (mb >> gg) & 1u;
            const float  s0  = on ? fmaf(t0, kvv.x, -m0) : -1e30f;
            const float  s1  = on ? fmaf(t1, kvv.x, -m1) : -1e30f;
            const float  e0  = EXP2F(s0);                 // v_exp_f32
            const float  e1  = EXP2F(s1);
            den0 += e0;
            num0  = fmaf(e0, kvv.y, num0);
            den1 += e1;
            num1  = fmaf(e1, kvv.y, num1);
        }
    }

    out[(size_t)b * FF + fa] = num0 / den0;
    out[(size_t)b * FF + fb] = num1 / den1;
}

// ---------------------------------------------------------------------------
// Launch: GEMM (Q,K,V into workspace) then fused attention. Workspace use:
// 3 * 64 * 2048 * 4 B = 1.5 MB.
// ---------------------------------------------------------------------------
extern "C" void kernel_launch(void* const* d_in, const int* in_sizes, int n_in,
                              void* d_out, int out_size, void* d_ws, size_t ws_size,
                              hipStream_t stream)
{
    const float* x     = (const float*)d_in[0];
    const int*   fmask = (const int*)  d_in[1];
    const float* Wq    = (const float*)d_in[2];
    const float* Wk    = (const float*)d_in[3];
    const float* Wv    = (const float*)d_in[4];

    float* Qv = (float*)d_ws;
    float* Kv = Qv + (size_t)BB * FF;
    float* Vv = Kv + (size_t)BB * FF;

    dim3 g1(FF / 16, BB / 16, 3);       // (128, 4, 3) tiles
    qkv_wmma_gemm<<<g1, 32, 0, stream>>>(x, Wq, Wk, Wv, Qv, Kv, Vv);

    dim3 g2(FF / 512, BB);              // (4, 64) blocks
    feature_attn<<<g2, 256, 0, stream>>>(Qv, Kv, Vv, fmask, (float*)d_out);
}